// MultiviewFeatureFusion_32504312496783
// MI455X (gfx1250) — compile-verified
//
#include <hip/hip_runtime.h>
#include <hip/hip_bf16.h>

// MultiviewFeatureFusion for MI455X (gfx1250, wave32).
//
// fusion[r, c] = sum_{k=0..3} features[4r+k, c]   (features: 32768 x 2048 f32)
// fusion_pids[r] = (float)pids[4r]                (8192 values)
//
// Memory-bound (≈335 MB moved @ 23.3 TB/s ≈ 14.4 us floor). The row-group sum
// is expressed as S·F with a 0/1 selection matrix and executed on the CDNA5
// matrix pipe: 16 chained V_WMMA_F32_16X16X4_F32 per 16x16 output tile, with a
// one-hot A matrix per K-step. Keeps VALU free; precision stays f32.

typedef __attribute__((ext_vector_type(2))) float v2f;
typedef __attribute__((ext_vector_type(8))) float v8f;

#define C_DIM      2048      // channels
#define N_ROWS     32768     // input rows
#define VIEW_NUM   4
#define FUSE_ROWS  (N_ROWS / VIEW_NUM)          // 8192
#define COL_TILES  (C_DIM / 16)                 // 128
#define ROW_TILES  (FUSE_ROWS / 16)             // 512
#define WAVES_PER_BLOCK 8

__global__ __launch_bounds__(32 * WAVES_PER_BLOCK)
void mvff_wmma_fuse_kernel(const float* __restrict__ features,
                           float* __restrict__ out)
{
    // One wave per 16x16 output tile. Consecutive waves in a block cover
    // consecutive 16-column tiles of the same 16-fusion-row group, so the
    // block streams a contiguous 64-row x 128-col input panel (good L2 reuse).
    const int wave    = threadIdx.x >> 5;
    const int lane    = threadIdx.x & 31;
    const int tileIdx = blockIdx.x * WAVES_PER_BLOCK + wave;
    const int tile_c  = tileIdx & (COL_TILES - 1);   // 0..127
    const int tile_r  = tileIdx >> 7;                // 0..511

    const int n = lane & 15;   // column within tile (N index)
    const int h = lane >> 4;   // half-wave selector

    const long long c0        = (long long)tile_c * 16;
    const long long inRowBase = (long long)tile_r * 64;   // 64 input rows / tile

    // Loop-invariant per-lane column pointer for B loads.
    const float* colPtr = features + c0 + n;

    v8f acc = {};   // C/D: 16x16 f32 accumulator, 8 VGPRs

    #pragma unroll
    for (int t = 0; t < 16; ++t) {
        // B (4x16 f32, 2 VGPRs): v0 -> K = 2h, v1 -> K = 2h+1, N = lane&15.
        const long long r0 = inRowBase + 4 * t + 2 * h;
        v2f b;
        b.x = colPtr[r0 * (long long)C_DIM];
        b.y = colPtr[(r0 + 1) * (long long)C_DIM];

        // A (16x4 f32, 2 VGPRs): one-hot row t -> A[m][k] = (m == t).
        // Lane layout: M = lane&15, v0 -> K = 2h, v1 -> K = 2h+1, so both
        // components carry the same one-hot value.
        const float av = (n == t) ? 1.0f : 0.0f;
        v2f a;
        a.x = av;
        a.y = av;

        // D = A x B + C  (accumulate the t-th group of 4 input rows into row t)
        acc = __builtin_amdgcn_wmma_f32_16x16x4_f32(
            /*neg_a=*/false, a, /*neg_b=*/false, b,
            /*c_mod=*/(short)0, acc, /*reuse_a=*/false, /*reuse_b=*/false);
    }

    // D layout: VGPR v -> fusion row (tile_r*16 + v + 8h), col (c0 + n).
    float* o = out + ((long long)tile_r * 16 + 8 * h) * (long long)C_DIM + c0 + n;
    #pragma unroll
    for (int v = 0; v < 8; ++v) {
        o[(long long)v * C_DIM] = acc[v];
    }
}

__global__ __launch_bounds__(256)
void mvff_pids_kernel(const int* __restrict__ pids,
                      float* __restrict__ out_pids)
{
    const int i = blockIdx.x * blockDim.x + threadIdx.x;
    if (i < FUSE_ROWS) {
        out_pids[i] = (float)pids[i * VIEW_NUM];
    }
}

extern "C" void kernel_launch(void* const* d_in, const int* in_sizes, int n_in,
                              void* d_out, int out_size, void* d_ws, size_t ws_size,
                              hipStream_t stream) {
    (void)in_sizes; (void)n_in; (void)d_ws; (void)ws_size; (void)out_size;

    const float* features = (const float*)d_in[0];
    const int*   pids     = (const int*)d_in[1];

    float* out_features = (float*)d_out;                              // 8192*2048
    float* out_pids     = (float*)d_out + (long long)FUSE_ROWS * C_DIM; // 8192

    // 512 row-tiles * 128 col-tiles = 65536 tiles, 8 waves (tiles) per block.
    const int totalTiles = ROW_TILES * COL_TILES;
    const int blocks     = totalTiles / WAVES_PER_BLOCK;   // 8192, exact
    mvff_wmma_fuse_kernel<<<blocks, 32 * WAVES_PER_BLOCK, 0, stream>>>(features, out_features);

    mvff_pids_kernel<<<(FUSE_ROWS + 255) / 256, 256, 0, stream>>>(pids, out_pids);
}